// EuclideanEmbedding_28003186770018
// MI455X (gfx1250) — compile-verified
//
#include <hip/hip_runtime.h>
#include <stdint.h>

#define N_EDGES    1600000
#define N_NODES    50000
#define D_SH       16
#define TILE_EDGES 256
#define BLOCK      256
#define NTILES     (N_EDGES / TILE_EDGES)   // 6250 exactly
#define NBLOCKS    625                       // 10 tiles per block
#define SH_TILE_BYTES (TILE_EDGES * D_SH * 4)   // 16 KB per buffer
#define QUAD_STRIDE   (TILE_EDGES * 16)          // 4096 B per quad plane

// ---- gfx1250 async global->LDS copy (ASYNCcnt path), inline asm form ----
// GVS addressing: mem_addr = SGPR64 + VGPR32_offset ; LDS dest addr in VGPR.
// Non-temporal hint: stream is read-once; keep the 3.2 MB atomic accumulator
// resident in L2 instead.
__device__ __forceinline__ void async_copy_b128_nt(unsigned lds_byte_addr,
                                                   const char* gbase_uniform,
                                                   unsigned voff_bytes) {
    asm volatile("global_load_async_to_lds_b128 %0, %1, %2 th:TH_LOAD_NT"
                 :: "v"(lds_byte_addr), "v"(voff_bytes), "s"(gbase_uniform)
                 : "memory");
}

__device__ __forceinline__ void wait_async_le4() {
    asm volatile("s_wait_asynccnt 0x4" ::: "memory");
}
__device__ __forceinline__ void wait_async_0() {
    asm volatile("s_wait_asynccnt 0x0" ::: "memory");
}

__device__ __forceinline__ void atomic_add_f32(float* p, float v) {
    unsafeAtomicAdd(p, v);   // native global_atomic_add_f32 (no return)
}

// Issue one 256-edge (16 KB) sh tile into an LDS buffer, quad-transposed:
// quad plane q at lds_buf + q*4096, edge e at +e*16. 4 b128 DMAs per lane.
__device__ __forceinline__ void issue_sh_tile(const char* sh_bytes, int tile,
                                              unsigned lds_buf, int t) {
    const char* gbase = sh_bytes + (unsigned long long)tile * (TILE_EDGES * 64);
#pragma unroll
    for (int k = 0; k < 4; ++k) {
        int qi = t + k * BLOCK;            // 0..1023 quad index in tile
        int q  = qi & 3;
        int e  = qi >> 2;
        unsigned voff = (unsigned)qi * 16u;                         // global byte off
        unsigned lds  = lds_buf + (unsigned)(q * QUAD_STRIDE + e * 16);
        async_copy_b128_nt(lds, gbase, voff);
    }
}

__global__ __launch_bounds__(BLOCK)
void euclid_scatter_kernel(const float* __restrict__ sh,
                           const float* __restrict__ cut,
                           const long long* __restrict__ recv,
                           const float* __restrict__ inv_p,
                           float* __restrict__ out) {
    __shared__ __align__(16) char smem[2 * SH_TILE_BYTES];
    const int t = threadIdx.x;
    const float inv = inv_p[0];
    const char* sh_bytes = (const char*)sh;
    // Generic shared pointer low 32 bits == LDS byte address (aperture layout).
    const unsigned lbase = (unsigned)(uintptr_t)(void*)smem;

    int tile = (int)blockIdx.x;
    if (tile >= NTILES) return;

    // Prologue: stage tile -> buf0, prefetch this tile's scalars into regs.
    int buf = 0;
    issue_sh_tile(sh_bytes, tile, lbase, t);
    float     c_cur = __builtin_nontemporal_load(&cut[(size_t)tile * TILE_EDGES + t]);
    long long r_cur = __builtin_nontemporal_load(&recv[(size_t)tile * TILE_EDGES + t]);

    while (true) {
        const int next_tile = tile + NBLOCKS;
        const bool has_next = (next_tile < NTILES);
        float c_nxt = 0.0f;
        long long r_nxt = 0;
        if (has_next) {
            issue_sh_tile(sh_bytes, next_tile, lbase + (unsigned)((buf ^ 1) * SH_TILE_BYTES), t);
            c_nxt = __builtin_nontemporal_load(&cut[(size_t)next_tile * TILE_EDGES + t]);
            r_nxt = __builtin_nontemporal_load(&recv[(size_t)next_tile * TILE_EDGES + t]);
            wait_async_le4();   // oldest 4 DMAs (current tile) complete; next 4 in flight
        } else {
            wait_async_0();
        }
        __syncthreads();        // all waves' DMA data visible in LDS

        // Process current tile: thread t owns edge (tile*256 + t).
        {
            const char* lb = smem + buf * SH_TILE_BYTES;
            const float w = c_cur * inv;
            float* o = out + (size_t)((int)r_cur) * D_SH;
#pragma unroll
            for (int q = 0; q < 4; ++q) {
                float4 v = *(const float4*)(lb + q * QUAD_STRIDE + t * 16);
                atomic_add_f32(o + q * 4 + 0, v.x * w);
                atomic_add_f32(o + q * 4 + 1, v.y * w);
                atomic_add_f32(o + q * 4 + 2, v.z * w);
                atomic_add_f32(o + q * 4 + 3, v.w * w);
            }
        }

        if (!has_next) break;
        __syncthreads();        // protect buf from being overwritten by next issue
        tile  = next_tile;
        buf  ^= 1;
        c_cur = c_nxt;
        r_cur = r_nxt;
    }
}

extern "C" void kernel_launch(void* const* d_in, const int* in_sizes, int n_in,
                              void* d_out, int out_size, void* d_ws, size_t ws_size,
                              hipStream_t stream) {
    (void)in_sizes; (void)n_in; (void)d_ws; (void)ws_size;
    const float*     sh   = (const float*)d_in[0];
    const float*     cut  = (const float*)d_in[1];
    const long long* recv = (const long long*)d_in[2];
    const float*     inv  = (const float*)d_in[3];
    float* out = (float*)d_out;

    // d_out is poisoned; atomics accumulate, so zero it first (capture-safe).
    hipMemsetAsync(d_out, 0, (size_t)out_size * sizeof(float), stream);
    euclid_scatter_kernel<<<NBLOCKS, BLOCK, 0, stream>>>(sh, cut, recv, inv, out);
}